// CLF_Adam_Layer_60258391163408
// MI455X (gfx1250) — compile-verified
//
#include <hip/hip_runtime.h>

typedef __attribute__((ext_vector_type(2))) float v2f;
typedef __attribute__((ext_vector_type(8))) float v8f;

#define D_FEAT    64
#define L2_FEAT   (64 * 64)
#define L3_FEAT   (64 * 64 * 64)
#define SEG_FEATS (D_FEAT + L2_FEAT + L3_FEAT)   // 266304
#define L3_BASE   (D_FEAT + L2_FEAT)             // 4160

// One wave (32 threads) per block. blockIdx.x = segment (b*32+s), 0..255.
// blockIdx.y = 0        : store level1 (-x) and level2 (x (x)T via WMMA)
// blockIdx.y = 1..16    : store level3 rows k = 4*(y-1) .. 4*(y-1)+3
__global__ __launch_bounds__(32)
void clf_signature_kernel(const float* __restrict__ x, float* __restrict__ out) {
  const int seg  = blockIdx.x;
  const int part = blockIdx.y;
  const int lane = threadIdx.x;
  const int l15  = lane & 15;
  const bool lo  = lane < 16;

  const float* xs = x + (size_t)seg * D_FEAT;
  float* os = out + (size_t)seg * SEG_FEATS;

  // x-chunk values for the four 16-wide chunks; lanes 0-15 carry the data.
  float xr0 = xs[0 * 16 + l15];
  float xr1 = xs[1 * 16 + l15];
  float xr2 = xs[2 * 16 + l15];
  float xr3 = xs[3 * 16 + l15];

  // WMMA 16x16x4 f32 fragments: put chunk in K=0 (lanes 0-15, VGPR0),
  // zero everywhere else so only the rank-1 term contributes.
  v2f f[4];
  f[0].x = lo ? xr0 : 0.0f; f[0].y = 0.0f;
  f[1].x = lo ? xr1 : 0.0f; f[1].y = 0.0f;
  f[2].x = lo ? xr2 : 0.0f; f[2].y = 0.0f;
  f[3].x = lo ? xr3 : 0.0f; f[3].y = 0.0f;

  // D-matrix layout: VGPR r -> row M = r + (lane<16 ? 0 : 8), col N = lane%16.
  const int mbase = (lo ? 0 : 8) * 64 + l15;

  if (part == 0) {
    // ---- level 1: v = -x (2 elements per lane) ----
    __builtin_nontemporal_store(-xs[lane],      os + lane);
    __builtin_nontemporal_store(-xs[32 + lane], os + 32 + lane);
    // ---- level 2: x (x)T, sixteen 16x16 WMMA tiles ----
#pragma unroll
    for (int tr = 0; tr < 4; ++tr) {
#pragma unroll
      for (int tc = 0; tc < 4; ++tc) {
        v8f acc = {};
        acc = __builtin_amdgcn_wmma_f32_16x16x4_f32(
            false, f[tr], false, f[tc], (short)0, acc, false, false);
        float* tb = os + D_FEAT + (tr * 16) * 64 + tc * 16;
#pragma unroll
        for (int r = 0; r < 8; ++r)
          __builtin_nontemporal_store(acc[r], tb + mbase + r * 64);
      }
    }
  } else {
    // ---- level 3: out3[k, j, m] = (-x[k]) * level2[j, m] ----
    const int k0 = (part - 1) * 4;
    const float s0 = -xs[k0 + 0];   // uniform -> scalar loads
    const float s1 = -xs[k0 + 1];
    const float s2 = -xs[k0 + 2];
    const float s3 = -xs[k0 + 3];
#pragma unroll
    for (int tr = 0; tr < 4; ++tr) {
#pragma unroll
      for (int tc = 0; tc < 4; ++tc) {
        v8f acc = {};
        acc = __builtin_amdgcn_wmma_f32_16x16x4_f32(
            false, f[tr], false, f[tc], (short)0, acc, false, false);
        float* tb = os + L3_BASE + (tr * 16) * 64 + tc * 16;
#pragma unroll
        for (int r = 0; r < 8; ++r) {
          const float a = acc[r];
          const int o = mbase + r * 64;
          __builtin_nontemporal_store(s0 * a, tb + (k0 + 0) * 4096 + o);
          __builtin_nontemporal_store(s1 * a, tb + (k0 + 1) * 4096 + o);
          __builtin_nontemporal_store(s2 * a, tb + (k0 + 2) * 4096 + o);
          __builtin_nontemporal_store(s3 * a, tb + (k0 + 3) * 4096 + o);
        }
      }
    }
  }
}

extern "C" void kernel_launch(void* const* d_in, const int* in_sizes, int n_in,
                              void* d_out, int out_size, void* d_ws, size_t ws_size,
                              hipStream_t stream) {
  const float* x = (const float*)d_in[0];
  float* out = (float*)d_out;
  const int n_segments = in_sizes[0] / D_FEAT;  // B * N_SEGMENTS = 256
  dim3 grid(n_segments, 17);
  dim3 block(32);
  clf_signature_kernel<<<grid, block, 0, stream>>>(x, out);
}